// KnowledgeModule_57535381897728
// MI455X (gfx1250) — compile-verified
//
#include <hip/hip_runtime.h>
#include <math.h>

// ---------------------------------------------------------------------------
// Sparse circuit forward (log semiring), 4 layers:
//   L0: segmented sum (1,000,000 outs), L1: segmented LSE (500,000),
//   L2: segmented sum (250,000),        L3: segmented LSE (100,000 -> d_out)
// HBM-bound (AI ~ 0.1 FLOP/B, roofline ~6-10us at 23.3 TB/s). Optimizations:
//  - row pointers from sorted csr via boundary detect (+ wave32 shfl_up)
//  - per-block edge window staged into LDS with gfx1250 async global->LDS
//    (B128 chunks from a 16B-aligned window base, B32 tail)
//  - NT loads on single-use streams, hot value tables stay cacheable
// ---------------------------------------------------------------------------

static constexpr int N_OUT0 = 1000000;
static constexpr int N_OUT1 = 500000;
static constexpr int N_OUT2 = 250000;
static constexpr int N_OUT3 = 100000;
static constexpr int MAX_FANIN = 14;          // rng.randint(2, 15)
static constexpr int BLOCK = 256;             // 8 wave32 waves
static constexpr int WIN_MAX = BLOCK * MAX_FANIN + 4; // +4: 16B base alignment slack
static constexpr float LSE_EPS = 1e-15f;
static constexpr float NEG_CUTOFF = -0.69314718055994530942f; // -ln 2

#if defined(__HIP_DEVICE_COMPILE__) && __has_builtin(__builtin_amdgcn_global_load_async_to_lds_b32)
#define HAVE_ASYNC_LDS 1
#else
#define HAVE_ASYNC_LDS 0
#endif

#if defined(__HIP_DEVICE_COMPILE__) && __has_builtin(__builtin_amdgcn_global_load_async_to_lds_b128)
#define HAVE_ASYNC_LDS_B128 1
#else
#define HAVE_ASYNC_LDS_B128 0
#endif

#if HAVE_ASYNC_LDS
typedef __attribute__((address_space(1))) int gbl_int;   // global (AS1)
typedef __attribute__((address_space(3))) int lds_int;   // LDS (AS3)
#endif
#if HAVE_ASYNC_LDS_B128
typedef int v4i __attribute__((ext_vector_type(4)));
typedef __attribute__((address_space(1))) v4i gbl_v4i;
typedef __attribute__((address_space(3))) v4i lds_v4i;
#endif

// gsrc and lds_dst are both 16-byte aligned; copies ndwords dwords.
__device__ __forceinline__ void async_window_copy(const int* __restrict__ gsrc,
                                                  int* __restrict__ lds_dst,
                                                  int ndwords) {
#if HAVE_ASYNC_LDS
#if HAVE_ASYNC_LDS_B128
  int nchunks = ndwords >> 2;                 // full 16B chunks
  for (int c = threadIdx.x; c < nchunks; c += BLOCK) {
    __builtin_amdgcn_global_load_async_to_lds_b128(
        (gbl_v4i*)(gsrc + 4 * c), (lds_v4i*)(lds_dst + 4 * c), 0, 0);
  }
  for (int t = (nchunks << 2) + threadIdx.x; t < ndwords; t += BLOCK) {
    __builtin_amdgcn_global_load_async_to_lds_b32(
        (gbl_int*)(gsrc + t), (lds_int*)(lds_dst + t), 0, 0);
  }
#else
  for (int t = threadIdx.x; t < ndwords; t += BLOCK) {
    __builtin_amdgcn_global_load_async_to_lds_b32(
        (gbl_int*)(gsrc + t), (lds_int*)(lds_dst + t), 0, 0);
  }
#endif
#if __has_builtin(__builtin_amdgcn_s_wait_asynccnt)
  __builtin_amdgcn_s_wait_asynccnt(0);
#else
  asm volatile("s_wait_asynccnt 0x0" ::: "memory");
#endif
#else
  for (int t = threadIdx.x; t < ndwords; t += BLOCK) {
    lds_dst[t] = __builtin_nontemporal_load(gsrc + t);
  }
#endif
  __syncthreads();
}

// x0 = [-inf, 0] ++ interleave(w, log1mexp(w))
__global__ void encode_input_kernel(const float* __restrict__ w,
                                    float* __restrict__ x0, int n) {
  int j = blockIdx.x * blockDim.x + threadIdx.x;
  if (j == 0) { x0[0] = -INFINITY; x0[1] = 0.0f; }
  if (j < n) {
    float wi = w[j];
    // log(1 - exp(wi)), Maechler 2012, eps = 0
    float neg = (wi > NEG_CUTOFF) ? logf(-expm1f(wi))
                                  : log1pf(-expf(wi));
    x0[2 + 2 * j] = wi;
    x0[3 + 2 * j] = neg;
  }
}

// csr is sorted segment ids (np.repeat(arange, lengths)); every segment has
// length >= 2, so boundary detection fully populates rs[0..n_out-1].
// Previous element comes from the neighboring lane (wave32 shfl) so the
// stream is loaded exactly once per element.
__global__ void row_starts_kernel(const int* __restrict__ csr, int n_edges,
                                  int* __restrict__ rs, int n_out) {
  int e = blockIdx.x * blockDim.x + threadIdx.x;
  if (e < n_edges) {
    int s  = __builtin_nontemporal_load(csr + e);       // single-use stream: NT
    int sp = __shfl_up(s, 1);                            // lane L-1's value
    if ((threadIdx.x & 31) == 0)
      sp = (e == 0) ? -1 : __builtin_nontemporal_load(csr + e - 1);
    if (s != sp) rs[s] = e;
  } else if (e == n_edges) {
    rs[n_out] = n_edges;
  }
}

// One thread per output segment, fan-in in [2,14]. The block's 256 segments
// cover ONE contiguous window of ptrs (<= 3584 dwords): stage it into LDS
// with coalesced async copies, then do the per-segment reads from LDS.
__global__ __launch_bounds__(BLOCK)
void seg_sum_kernel(const float* __restrict__ xin,
                    const int* __restrict__ ptrs,
                    const int* __restrict__ rs,
                    float* __restrict__ xout, int n_out) {
  __shared__ __attribute__((aligned(16))) int win[WIN_MAX];
  __shared__ int sh_begin, sh_end;

  int i0 = blockIdx.x * BLOCK;
  int i  = i0 + threadIdx.x;
  int iclamp = min(i, n_out - 1);
  int s = rs[iclamp];
  int e = rs[iclamp + 1];
  int last_valid = min(i0 + BLOCK, n_out) - 1;
  if (threadIdx.x == 0) sh_begin = s;
  if (i == last_valid)  sh_end   = e;
  __syncthreads();
  int wb_al = sh_begin & ~3;            // 16B-aligned window base (still in-bounds)
  async_window_copy(ptrs + wb_al, win, sh_end - wb_al);

  if (i >= n_out) return;
  int base = s - wb_al;
  int len  = e - s;
  float acc = 0.0f;
#pragma unroll
  for (int k = 0; k < MAX_FANIN; ++k) {
    if (k < len) acc += xin[win[base + k]];
  }
  xout[i] = acc;
}

// Segmented logsumexp with max-shift: out = log(sum exp(g - m) + eps) + m.
// Gathered values cached in registers (constant-indexed unrolled array) so
// the xin gather happens exactly once per edge.
__global__ __launch_bounds__(BLOCK)
void seg_lse_kernel(const float* __restrict__ xin,
                    const int* __restrict__ ptrs,
                    const int* __restrict__ rs,
                    float* __restrict__ xout, int n_out) {
  __shared__ __attribute__((aligned(16))) int win[WIN_MAX];
  __shared__ int sh_begin, sh_end;

  int i0 = blockIdx.x * BLOCK;
  int i  = i0 + threadIdx.x;
  int iclamp = min(i, n_out - 1);
  int s = rs[iclamp];
  int e = rs[iclamp + 1];
  int last_valid = min(i0 + BLOCK, n_out) - 1;
  if (threadIdx.x == 0) sh_begin = s;
  if (i == last_valid)  sh_end   = e;
  __syncthreads();
  int wb_al = sh_begin & ~3;
  async_window_copy(ptrs + wb_al, win, sh_end - wb_al);

  if (i >= n_out) return;
  int base = s - wb_al;
  int len  = e - s;
  float g[MAX_FANIN];
  float m = -INFINITY;
#pragma unroll
  for (int k = 0; k < MAX_FANIN; ++k) {
    float v = -INFINITY;
    if (k < len) v = xin[win[base + k]];
    g[k] = v;
    m = fmaxf(m, v);
  }
  float acc = 0.0f;
#pragma unroll
  for (int k = 0; k < MAX_FANIN; ++k) {
    if (k < len) acc += expf(g[k] - m);
  }
  xout[i] = logf(acc + LSE_EPS) + m;
}

extern "C" void kernel_launch(void* const* d_in, const int* in_sizes, int n_in,
                              void* d_out, int out_size, void* d_ws, size_t ws_size,
                              hipStream_t stream) {
  // setup_inputs() dict order: ptrs0,csr0,ptrs1,csr1,ptrs2,csr2,ptrs3,csr3,weights
  const int*   ptrs0 = (const int*)d_in[0];
  const int*   csr0  = (const int*)d_in[1];
  const int*   ptrs1 = (const int*)d_in[2];
  const int*   csr1  = (const int*)d_in[3];
  const int*   ptrs2 = (const int*)d_in[4];
  const int*   csr2  = (const int*)d_in[5];
  const int*   ptrs3 = (const int*)d_in[6];
  const int*   csr3  = (const int*)d_in[7];
  const float* w     = (const float*)d_in[8];

  const int nb_vars = in_sizes[8];
  const int ne0 = in_sizes[0];
  const int ne1 = in_sizes[2];
  const int ne2 = in_sizes[4];
  const int ne3 = in_sizes[6];

  // Carve workspace (all buffers fully overwritten each call; no cross-call state)
  char*  ws  = (char*)d_ws;
  size_t off = 0;
  auto alloc = [&](size_t bytes) -> void* {
    void* p = ws + off;
    off += (bytes + 255) & ~size_t(255);
    return p;
  };
  float* x0  = (float*)alloc((size_t)(2 + 2 * nb_vars) * sizeof(float));
  float* x1  = (float*)alloc((size_t)N_OUT0 * sizeof(float));
  float* x2  = (float*)alloc((size_t)N_OUT1 * sizeof(float));
  float* x3  = (float*)alloc((size_t)N_OUT2 * sizeof(float));
  int*   rs0 = (int*)alloc(((size_t)N_OUT0 + 1) * sizeof(int));
  int*   rs1 = (int*)alloc(((size_t)N_OUT1 + 1) * sizeof(int));
  int*   rs2 = (int*)alloc(((size_t)N_OUT2 + 1) * sizeof(int));
  int*   rs3 = (int*)alloc(((size_t)N_OUT3 + 1) * sizeof(int));
  (void)ws_size; (void)n_in; (void)out_size;

  auto cdiv = [](int a, int b) { return (a + b - 1) / b; };

  encode_input_kernel<<<cdiv(nb_vars, BLOCK), BLOCK, 0, stream>>>(w, x0, nb_vars);

  // Layer 0: ProdLayer (segmented sum in log space)
  row_starts_kernel<<<cdiv(ne0 + 1, BLOCK), BLOCK, 0, stream>>>(csr0, ne0, rs0, N_OUT0);
  seg_sum_kernel<<<cdiv(N_OUT0, BLOCK), BLOCK, 0, stream>>>(x0, ptrs0, rs0, x1, N_OUT0);

  // Layer 1: LogSumLayer (segmented logsumexp)
  row_starts_kernel<<<cdiv(ne1 + 1, BLOCK), BLOCK, 0, stream>>>(csr1, ne1, rs1, N_OUT1);
  seg_lse_kernel<<<cdiv(N_OUT1, BLOCK), BLOCK, 0, stream>>>(x1, ptrs1, rs1, x2, N_OUT1);

  // Layer 2: ProdLayer
  row_starts_kernel<<<cdiv(ne2 + 1, BLOCK), BLOCK, 0, stream>>>(csr2, ne2, rs2, N_OUT2);
  seg_sum_kernel<<<cdiv(N_OUT2, BLOCK), BLOCK, 0, stream>>>(x2, ptrs2, rs2, x3, N_OUT2);

  // Layer 3: LogSumLayer -> final output
  row_starts_kernel<<<cdiv(ne3 + 1, BLOCK), BLOCK, 0, stream>>>(csr3, ne3, rs3, N_OUT3);
  seg_lse_kernel<<<cdiv(N_OUT3, BLOCK), BLOCK, 0, stream>>>(x3, ptrs3, rs3, (float*)d_out, N_OUT3);
}